// ProtoMulHeadAttn_73048803770585
// MI455X (gfx1250) — compile-verified
//
#include <hip/hip_runtime.h>
#include <hip/hip_bf16.h>

#define WAY   5
#define SHOT  5
#define HEADS 12
#define DIM   768
#define L     512
#define HD    64
#define SCALE 0.125f   // 1/sqrt(64)

typedef float v2f __attribute__((ext_vector_type(2)));
typedef float v8f __attribute__((ext_vector_type(8)));

// D = A(16x4,f32) * B(4x16,f32) + C(16x16,f32)  — CDNA5 wave32 WMMA
__device__ __forceinline__ v8f wmma_f32_4(v2f a, v2f b, v8f c) {
    return __builtin_amdgcn_wmma_f32_16x16x4_f32(
        /*neg_a=*/false, a, /*neg_b=*/false, b,
        /*c_mod=*/(short)0, c, /*reuse_a=*/false, /*reuse_b=*/false);
}

// monotone float<->uint encoding so unsigned atomicMax == float max
__device__ __forceinline__ unsigned fenc(float f) {
    unsigned u = __float_as_uint(f);
    return (u & 0x80000000u) ? ~u : (u | 0x80000000u);
}
__device__ __forceinline__ float fdec(unsigned e) {
    return __uint_as_float((e & 0x80000000u) ? (e & 0x7fffffffu) : ~e);
}

// ---------------------------------------------------------------------------
// Kernel 1: QKV projection  C[12800,768] = x[12800,768] @ W^T + b
// grid (400, 3, 3); block 256 (8 waves as 2x4); wave -> 16x64 strip
// (4 accumulators, A fragment loaded once per K-step -> 5 loads / 4 WMMAs)
// ---------------------------------------------------------------------------
__global__ __launch_bounds__(256) void qkv_proj(
    const float* __restrict__ x,
    const float* __restrict__ Wq, const float* __restrict__ bq,
    const float* __restrict__ Wk, const float* __restrict__ bk,
    const float* __restrict__ Wv, const float* __restrict__ bv,
    float* __restrict__ Qo, float* __restrict__ Ko, float* __restrict__ Vo)
{
    const float* W; const float* b; float* O;
    if (blockIdx.z == 0)      { W = Wq; b = bq; O = Qo; }
    else if (blockIdx.z == 1) { W = Wk; b = bk; O = Ko; }
    else                      { W = Wv; b = bv; O = Vo; }

    const int wave = threadIdx.x >> 5;
    const int lane = threadIdx.x & 31;
    const int half = lane >> 4;
    const int lr   = lane & 15;

    const int row0 = blockIdx.x * 32 + (wave >> 2) * 16;
    const int col0 = blockIdx.y * 256 + (wave & 3) * 64;

    const float* arow = x + (row0 + lr) * DIM + 2 * half;
    const float* brow = W + (col0 + lr) * DIM + 2 * half;   // B[k][n] = W[n][k]

    v8f c0 = {}, c1 = {}, c2 = {}, c3 = {};
    for (int k = 0; k < DIM; k += 4) {
        v2f a  = *(const v2f*)(arow + k);
        v2f b0 = *(const v2f*)(brow + k);
        v2f b1 = *(const v2f*)(brow + 16 * DIM + k);
        v2f b2 = *(const v2f*)(brow + 32 * DIM + k);
        v2f b3 = *(const v2f*)(brow + 48 * DIM + k);
        c0 = wmma_f32_4(a, b0, c0);
        c1 = wmma_f32_4(a, b1, c1);
        c2 = wmma_f32_4(a, b2, c2);
        c3 = wmma_f32_4(a, b3, c3);
    }

    float* orow = O + (row0 + 8 * half) * DIM + col0 + lr;
#pragma unroll
    for (int j = 0; j < 4; ++j) {
        const v8f acc  = (j == 0) ? c0 : (j == 1) ? c1 : (j == 2) ? c2 : c3;
        const float bj = b[col0 + j * 16 + lr];
#pragma unroll
        for (int g = 0; g < 8; ++g)
            orow[g * DIM + j * 16] = acc[g] + bj;
    }
}

// ---------------------------------------------------------------------------
// Kernel 2: branch-1 token attention, fused max-pool over l.
// grid (SHOT, HEADS, WAY); block 256 (8 waves).
// LDS: scores P[64,516] (~129 KB) + V slice Vs[512,72] (~144 KB) of 320 KB.
// ---------------------------------------------------------------------------
#define LP 516   // P row stride (floats): 516 % 64 == 4 -> conflict-free frags
#define VP 72    // Vs row stride: rows differ by 2 -> +16 banks between halves
__global__ __launch_bounds__(256) void attn_tok(
    const float* __restrict__ Q, const float* __restrict__ Km,
    const float* __restrict__ V, float* __restrict__ a1)
{
    __shared__ __align__(16) float P[64 * LP];
    __shared__ __align__(16) float Vs[L * VP];
    __shared__ unsigned smax[64];

    const int s = blockIdx.x, h = blockIdx.y, w = blockIdx.z;
    const int base = ((w * SHOT + s) * L) * DIM + h * HD;
    const float* q = Q  + base;
    const float* k = Km + base;
    const float* v = V  + base;

    const int wave = threadIdx.x >> 5;
    const int lane = threadIdx.x & 31;
    const int half = lane >> 4;
    const int lr   = lane & 15;

    // stage the whole V slice once (coalesced 256B rows)
    for (int i = threadIdx.x; i < L * HD; i += 256) {
        const int r = i >> 6, c = i & 63;
        Vs[r * VP + c] = v[r * DIM + c];
    }
    if (threadIdx.x < 64) smax[threadIdx.x] = fenc(-__builtin_inff());
    __syncthreads();

    for (int qt = 0; qt < L / 64; ++qt) {
        const int l0 = qt * 64;

        // ---- phase 1: S[64,512] = Q_tile @ K^T * SCALE
        // wave -> fixed 16-row strip (mi = wave>>1), 16 key tiles in groups
        // of 4 sharing one A fragment per K-step.
        {
            const int mi = wave >> 1;
            const float* ap = q + (l0 + mi * 16 + lr) * DIM + 2 * half;
            for (int g = 0; g < 4; ++g) {
                const int ni0 = (wave & 1) * 16 + g * 4;
                const float* bp = k + (ni0 * 16 + lr) * DIM + 2 * half;
                v8f c0 = {}, c1 = {}, c2 = {}, c3 = {};
#pragma unroll
                for (int kk = 0; kk < HD; kk += 4) {
                    v2f a  = *(const v2f*)(ap + kk);
                    v2f b0 = *(const v2f*)(bp + kk);
                    v2f b1 = *(const v2f*)(bp + 16 * DIM + kk);
                    v2f b2 = *(const v2f*)(bp + 32 * DIM + kk);
                    v2f b3 = *(const v2f*)(bp + 48 * DIM + kk);
                    c0 = wmma_f32_4(a, b0, c0);
                    c1 = wmma_f32_4(a, b1, c1);
                    c2 = wmma_f32_4(a, b2, c2);
                    c3 = wmma_f32_4(a, b3, c3);
                }
                float* pr = &P[(mi * 16 + 8 * half) * LP + ni0 * 16 + lr];
#pragma unroll
                for (int j = 0; j < 4; ++j) {
                    const v8f acc = (j == 0) ? c0 : (j == 1) ? c1 : (j == 2) ? c2 : c3;
#pragma unroll
                    for (int gg = 0; gg < 8; ++gg)
                        pr[gg * LP + j * 16] = acc[gg] * SCALE;
                }
            }
        }
        __syncthreads();

        // ---- phase 2: exact softmax, 4 lanes per row, 128 cols each
        {
            const int r = threadIdx.x >> 2, j = threadIdx.x & 3;
            float* row = &P[r * LP + j * 128];
            float m = -__builtin_inff();
            for (int i = 0; i < 128; ++i) m = fmaxf(m, row[i]);
            m = fmaxf(m, __shfl_xor(m, 1, 4));
            m = fmaxf(m, __shfl_xor(m, 2, 4));
            float sum = 0.f;
            for (int i = 0; i < 128; ++i) { float e = __expf(row[i] - m); row[i] = e; sum += e; }
            sum += __shfl_xor(sum, 1, 4);
            sum += __shfl_xor(sum, 2, 4);
            const float inv = 1.f / sum;
            for (int i = 0; i < 128; ++i) row[i] *= inv;
        }
        __syncthreads();

        // ---- phase 3: O[64,64] = P @ V (V from LDS), fused column-max.
        // wave -> mi = wave>>1, two N-tiles sharing one A fragment.
        {
            const int mi  = wave >> 1;
            const int ni0 = (wave & 1) * 2;
            const float* ap  = &P[(mi * 16 + lr) * LP + 2 * half];
            const float* vp0 = &Vs[(2 * half) * VP + ni0 * 16 + lr];
            const float* vp1 = vp0 + 16;
            v8f c0 = {}, c1 = {};
            for (int kk = 0; kk < L; kk += 4) {
                v2f a = *(const v2f*)(ap + kk);
                v2f b0, b1;
                b0.x = vp0[kk * VP];       b0.y = vp0[(kk + 1) * VP];
                b1.x = vp1[kk * VP];       b1.y = vp1[(kk + 1) * VP];
                c0 = wmma_f32_4(a, b0, c0);
                c1 = wmma_f32_4(a, b1, c1);
            }
#pragma unroll
            for (int j = 0; j < 2; ++j) {
                const v8f acc = (j == 0) ? c0 : c1;
                float mv = acc[0];
#pragma unroll
                for (int g = 1; g < 8; ++g) mv = fmaxf(mv, acc[g]);
                mv = fmaxf(mv, __shfl_xor(mv, 16, 32));  // fold the half-lanes
                if (half == 0) atomicMax(&smax[(ni0 + j) * 16 + lr], fenc(mv));
            }
        }
        __syncthreads();
    }

    if (threadIdx.x < 64)
        a1[(w * SHOT + s) * DIM + h * HD + threadIdx.x] = fdec(smax[threadIdx.x]);
}

// ---------------------------------------------------------------------------
// Kernel 3: branch-2 channel attention (5x5 scores) + fused l-max.
// grid (HD, HEADS, WAY); block 128. Block uniquely owns a2[w][*][h*64+d].
// ---------------------------------------------------------------------------
__global__ __launch_bounds__(128) void attn_chan(
    const float* __restrict__ Q, const float* __restrict__ Km,
    const float* __restrict__ V, float* __restrict__ a2)
{
    __shared__ float qd[SHOT][L], kd[SHOT][L], vd[SHOT][L];
    __shared__ float s2[SHOT][SHOT], w2[SHOT][SHOT];
    __shared__ float red[4];

    const int d = blockIdx.x, h = blockIdx.y, w = blockIdx.z;
    const int tid = threadIdx.x;

    for (int i = tid; i < SHOT * L; i += 128) {
        const int s = i >> 9, l = i & (L - 1);
        const int off = ((w * SHOT + s) * L + l) * DIM + h * HD + d;
        qd[s][l] = Q[off]; kd[s][l] = Km[off]; vd[s][l] = V[off];
    }
    __syncthreads();

    if (tid < SHOT * SHOT) {
        const int si = tid / SHOT, ti = tid % SHOT;
        float acc = 0.f;
        for (int l = 0; l < L; ++l) acc += qd[si][l] * kd[ti][l];
        s2[si][ti] = acc * SCALE;
    }
    __syncthreads();

    if (tid < SHOT) {
        float m = s2[tid][0];
        for (int t = 1; t < SHOT; ++t) m = fmaxf(m, s2[tid][t]);
        float e[SHOT]; float sum = 0.f;
        for (int t = 0; t < SHOT; ++t) { e[t] = __expf(s2[tid][t] - m); sum += e[t]; }
        for (int t = 0; t < SHOT; ++t) w2[tid][t] = e[t] / sum;
    }
    __syncthreads();

    for (int s = 0; s < SHOT; ++s) {
        float mx = -__builtin_inff();
        for (int l = tid; l < L; l += 128) {
            float val = 0.f;
            for (int t = 0; t < SHOT; ++t) val += w2[s][t] * vd[t][l];
            mx = fmaxf(mx, val);
        }
        for (int off = 16; off; off >>= 1) mx = fmaxf(mx, __shfl_xor(mx, off, 32));
        if ((tid & 31) == 0) red[tid >> 5] = mx;
        __syncthreads();
        if (tid == 0)
            a2[(w * SHOT + s) * DIM + h * HD + d] =
                fmaxf(fmaxf(red[0], red[1]), fmaxf(red[2], red[3]));
        __syncthreads();
    }
}

// ---------------------------------------------------------------------------
// block-wide sum reduction (256 threads, wave32)
// ---------------------------------------------------------------------------
__device__ __forceinline__ float blockSum(float v, float* scr) {
    for (int off = 16; off; off >>= 1) v += __shfl_xor(v, off, 32);
    if ((threadIdx.x & 31) == 0) scr[threadIdx.x >> 5] = v;
    __syncthreads();
    float r = (threadIdx.x < 8) ? scr[threadIdx.x] : 0.f;
    for (int off = 4; off; off >>= 1) r += __shfl_xor(r, off, 8);
    if (threadIdx.x == 0) scr[0] = r;
    __syncthreads();
    r = scr[0];
    __syncthreads();
    return r;
}

// ---------------------------------------------------------------------------
// Kernel 4: per-(w,s) layernorms, attn average, tanh.wp logit.
// grid 25; block 256.
// ---------------------------------------------------------------------------
__global__ __launch_bounds__(256) void head_part1(
    const float* __restrict__ a1, const float* __restrict__ a2,
    const float* __restrict__ gamma, const float* __restrict__ beta,
    const float* __restrict__ wp, const float* __restrict__ bp,
    float* __restrict__ attnbuf, float* __restrict__ logits)
{
    __shared__ float scr[8];
    const int ws = blockIdx.x, tid = threadIdx.x;
    const float* r1 = a1 + ws * DIM;
    const float* r2 = a2 + ws * DIM;

    float s1 = 0.f, q1 = 0.f, s2v = 0.f, q2 = 0.f;
    for (int i = tid; i < DIM; i += 256) {
        float v1 = r1[i]; s1 += v1; q1 += v1 * v1;
        float v2 = r2[i]; s2v += v2; q2 += v2 * v2;
    }
    s1  = blockSum(s1,  scr);
    q1  = blockSum(q1,  scr);
    s2v = blockSum(s2v, scr);
    q2  = blockSum(q2,  scr);

    const float m1 = s1 / DIM,  var1 = q1 / DIM - m1 * m1;
    const float m2 = s2v / DIM, var2 = q2 / DIM - m2 * m2;
    const float i1 = rsqrtf(var1 + 1e-5f);
    const float i2 = rsqrtf(var2 + 1e-5f);

    float dot = 0.f;
    for (int i = tid; i < DIM; i += 256) {
        const float g = gamma[i], bt = beta[i];
        const float ln1 = (r1[i] - m1) * i1 * g + bt;
        const float ln2 = (r2[i] - m2) * i2 * g + bt;
        const float at = 0.5f * (ln1 + ln2);
        attnbuf[ws * DIM + i] = at;
        dot += tanhf(at) * wp[i];
    }
    dot = blockSum(dot, scr);
    if (tid == 0) logits[ws] = dot + bp[0];
}

// ---------------------------------------------------------------------------
// Kernel 5: shot softmax + weighted sum + Wo projection + sum over shots.
// grid WAY; block 256.
// ---------------------------------------------------------------------------
__global__ __launch_bounds__(256) void head_part2(
    const float* __restrict__ attnbuf, const float* __restrict__ logits,
    const float* __restrict__ Wo, const float* __restrict__ bo,
    float* __restrict__ out)
{
    __shared__ float c[DIM];
    __shared__ float pw[SHOT];
    const int w = blockIdx.x, tid = threadIdx.x;

    if (tid == 0) {
        float m = -__builtin_inff();
        for (int s = 0; s < SHOT; ++s) m = fmaxf(m, logits[w * SHOT + s]);
        float e[SHOT]; float sum = 0.f;
        for (int s = 0; s < SHOT; ++s) { e[s] = __expf(logits[w * SHOT + s] - m); sum += e[s]; }
        for (int s = 0; s < SHOT; ++s) pw[s] = e[s] / sum;
    }
    __syncthreads();

    for (int d = tid; d < DIM; d += 256) {
        float acc = 0.f;
        for (int s = 0; s < SHOT; ++s) acc += pw[s] * attnbuf[(w * SHOT + s) * DIM + d];
        c[d] = acc;
    }
    __syncthreads();

    for (int n = tid; n < DIM; n += 256) {
        float acc = 0.f;
        const float* wr = Wo + n * DIM;
        for (int d = 0; d < DIM; ++d) acc += c[d] * wr[d];
        out[w * DIM + n] = acc + (float)SHOT * bo[n];   // bo added per-shot then summed
    }
}

// ---------------------------------------------------------------------------
extern "C" void kernel_launch(void* const* d_in, const int* in_sizes, int n_in,
                              void* d_out, int out_size, void* d_ws, size_t ws_size,
                              hipStream_t stream)
{
    (void)in_sizes; (void)n_in; (void)out_size; (void)ws_size;
    const float* x     = (const float*)d_in[0];
    const float* Wq    = (const float*)d_in[1];
    const float* bq    = (const float*)d_in[2];
    const float* Wk    = (const float*)d_in[3];
    const float* bk    = (const float*)d_in[4];
    const float* Wv    = (const float*)d_in[5];
    const float* bv    = (const float*)d_in[6];
    const float* gamma = (const float*)d_in[7];
    const float* beta  = (const float*)d_in[8];
    const float* wp    = (const float*)d_in[9];
    const float* bp    = (const float*)d_in[10];
    const float* Wo    = (const float*)d_in[11];
    const float* bo    = (const float*)d_in[12];
    float* out = (float*)d_out;

    const size_t NQKV = (size_t)WAY * SHOT * L * DIM;  // 9,830,400 floats
    float* ws      = (float*)d_ws;
    float* Qb      = ws;
    float* Kb      = ws + NQKV;
    float* Vb      = ws + 2 * NQKV;
    float* a1      = ws + 3 * NQKV;              // 25*768
    float* a2      = a1 + WAY * SHOT * DIM;      // 25*768
    float* attnbuf = a2 + WAY * SHOT * DIM;      // 25*768
    float* logits  = attnbuf + WAY * SHOT * DIM; // 25

    dim3 g1((WAY * SHOT * L) / 32, DIM / 256, 3);
    qkv_proj<<<g1, 256, 0, stream>>>(x, Wq, bq, Wk, bk, Wv, bv, Qb, Kb, Vb);

    dim3 g2(SHOT, HEADS, WAY);
    attn_tok<<<g2, 256, 0, stream>>>(Qb, Kb, Vb, a1);

    dim3 g3(HD, HEADS, WAY);
    attn_chan<<<g3, 128, 0, stream>>>(Qb, Kb, Vb, a2);

    head_part1<<<WAY * SHOT, 256, 0, stream>>>(a1, a2, gamma, beta, wp, bp, attnbuf, logits);
    head_part2<<<WAY, 256, 0, stream>>>(attnbuf, logits, Wo, bo, out);
}